// SetCriterionWeakSup_71803263255228
// MI455X (gfx1250) — compile-verified
//
#include <hip/hip_runtime.h>
#include <hip/hip_bf16.h>
#include <math.h>

// ---------------------------------------------------------------------------
// MI455X / gfx1250 implementation of SetCriterion weak-supervision loss.
// Memory-bound streaming problem (~168 MB @ 23.3 TB/s ~= 7.2us floor).
// Reductions use V_WMMA_F32_16X16X4_F32 (f32 matrix pipe) instead of LDS trees.
// ---------------------------------------------------------------------------

typedef float v2f __attribute__((ext_vector_type(2)));
typedef float v8f __attribute__((ext_vector_type(8)));

// Wave32 sum via f32 WMMA: A(16x4) holds lane values (lanes0-15 -> K0,
// lanes16-31 -> K2), B = ones => D[m][j] = v_m + v_{m+16}. Lane sums its
// 8 D-VGPRs (rows 0-7 or 8-15), then one shfl_xor(16) folds the halves.
__device__ __forceinline__ float wave_sum_wmma(float v) {
  v2f a; a[0] = v;    a[1] = 0.0f;
  v2f b; b[0] = 1.0f; b[1] = 1.0f;
  v8f c = {};
  c = __builtin_amdgcn_wmma_f32_16x16x4_f32(false, a, false, b,
                                            (short)0, c, false, false);
  float s = c[0] + c[1] + c[2] + c[3] + c[4] + c[5] + c[6] + c[7];
  s += __shfl_xor(s, 16, 32);
  return s;  // every lane holds the wave total
}

__device__ __forceinline__ float wave_max32(float v) {
  #pragma unroll
  for (int o = 16; o > 0; o >>= 1) v = fmaxf(v, __shfl_xor(v, o, 32));
  return v;
}

// Block-wide sum (256 threads = 8 waves). Result valid on thread 0.
__device__ __forceinline__ float block_sum_wmma(float v, float* scratch) {
  float s = wave_sum_wmma(v);
  int lane = threadIdx.x & 31, wid = threadIdx.x >> 5;
  if (lane == 0) scratch[wid] = s;
  __syncthreads();
  float r = 0.f;
  if (threadIdx.x == 0) {
    int nw = (int)((blockDim.x + 31) >> 5);
    for (int i = 0; i < nw; i++) r += scratch[i];
  }
  __syncthreads();
  return r;
}

__device__ __forceinline__ float block_max_f(float v, float* scratch) {
  float s = wave_max32(v);
  int lane = threadIdx.x & 31, wid = threadIdx.x >> 5;
  if (lane == 0) scratch[wid] = s;
  __syncthreads();
  float r = -3.4e38f;
  if (threadIdx.x == 0) {
    int nw = (int)((blockDim.x + 31) >> 5);
    for (int i = 0; i < nw; i++) r = fmaxf(r, scratch[i]);
  }
  __syncthreads();
  return r;
}

// stable log-sigmoid pair: lf = log_sigmoid(x) = min(x,0) - log1p(e^{-|x|});
// lb = log_sigmoid(-x) = lf - x.
__device__ __forceinline__ void lsig2(float xv, float& lf, float& lb) {
  float e = __expf(-fabsf(xv));
  lf = fminf(xv, 0.f) - __logf(1.f + e);
  lb = lf - xv;
}

#define HH 256
#define WW 256
#define NN 64
#define NCH 4          // column-pass row chunks
#define CHROWS (HH / NCH)

// --------------------------- K1: pairwise + row reductions ------------------
__global__ __launch_bounds__(256) void k1_pair_row(
    const float* __restrict__ src, const int* __restrict__ tb,
    const float* __restrict__ ts,
    float* __restrict__ pair_sim, float* __restrict__ pair_cnt,
    float* __restrict__ rowmax, float* __restrict__ rowor) {
  __shared__ float lf[3][264];
  __shared__ float lb[3][264];
  __shared__ float scratch[8];

  const int n = blockIdx.x >> 8;
  const int y = blockIdx.x & 255;
  const int x = threadIdx.x;

  // Stage 3 rows (y-2, y, y+2) of lfg/lbg with +-2 halo; OOB taps are 0
  // (reference zero-pads AFTER log_sigmoid).
  #pragma unroll
  for (int r = 0; r < 3; r++) {
    int yy = y + (r - 1) * 2;
    bool vr = (yy >= 0) && (yy < HH);
    float a = 0.f, bv = 0.f;
    if (vr) lsig2(src[((size_t)(n * HH) + yy) * WW + x], a, bv);
    lf[r][x + 2] = a; lb[r][x + 2] = bv;
    if (x < 4) {
      int c = (x < 2) ? (x - 2) : (254 + x);  // -2,-1,256,257
      float ah = 0.f, bh = 0.f;
      if (vr && c >= 0 && c < WW)
        lsig2(src[((size_t)(n * HH) + yy) * WW + c], ah, bh);
      lf[r][c + 2] = ah; lb[r][c + 2] = bh;
    }
  }
  __syncthreads();

  const float lfc = lf[1][x + 2];
  const float lbc = lb[1][x + 2];
  const float xv  = lfc - lbc;  // recover raw src value (x = lf - lb)
  const int   tbv = tb[((size_t)(n * HH) + y) * WW + x];
  const float tbf = tbv ? 1.f : 0.f;

  // unfold order (i-major, j-minor, center removed), offsets (i-1)*2,(j-1)*2
  const int rr[8] = {0, 0, 0, 1, 1, 2, 2, 2};
  const int cc[8] = {-2, 0, 2, -2, 2, -2, 0, 2};

  float asum = 0.f, acnt = 0.f;
  const float* tsp = ts + (((size_t)n * 8) * HH + y) * WW + x;
  #pragma unroll
  for (int p = 0; p < 8; p++) {
    float tsv = __builtin_nontemporal_load(tsp + (size_t)p * HH * WW);
    float a  = lfc + lf[rr[p]][x + 2 + cc[p]];
    float b2 = lbc + lb[rr[p]][x + 2 + cc[p]];
    float m  = fmaxf(a, b2);
    float ls = m + __logf(1.f + __expf(-fabsf(a - b2)));  // logaddexp(a,b)
    float t  = ((tsv >= 0.3f) && (tbv != 0)) ? 1.f : 0.f;
    asum = fmaf(t, -ls, asum);
    acnt += t;
  }

  float rs = block_sum_wmma(asum, scratch);
  float rc = block_sum_wmma(acnt, scratch);
  float rm = block_max_f(xv, scratch);
  float ro = block_max_f(tbf, scratch);
  if (threadIdx.x == 0) {
    pair_sim[blockIdx.x] = rs;
    pair_cnt[blockIdx.x] = rc;
    rowmax[n * HH + y] = rm;
    rowor[n * HH + y]  = ro;
  }
}

// --------------------------- K2: column max / or ----------------------------
__global__ __launch_bounds__(256) void k2_col(
    const float* __restrict__ src, const int* __restrict__ tb,
    float* __restrict__ colmax, float* __restrict__ color) {
  const int n = blockIdx.x, ch = blockIdx.y, x = threadIdx.x;
  const int y0 = ch * CHROWS;
  const float* sp = src + ((size_t)(n * HH) + y0) * WW + x;
  const int*   tp = tb  + ((size_t)(n * HH) + y0) * WW + x;
  float m = -3.4e38f, o = 0.f;
  #pragma unroll 4
  for (int i = 0; i < CHROWS; i++) {
    m = fmaxf(m, sp[(size_t)i * WW]);
    o = fmaxf(o, tp[(size_t)i * WW] ? 1.f : 0.f);
  }
  colmax[((size_t)(n * NCH) + ch) * WW + x] = m;
  color [((size_t)(n * NCH) + ch) * WW + x] = o;
}

// --------------------------- K3: projection dice ----------------------------
__global__ __launch_bounds__(256) void k3_proj(
    const float* __restrict__ rowmax, const float* __restrict__ rowor,
    const float* __restrict__ colmax, const float* __restrict__ color,
    float* __restrict__ proj) {
  __shared__ float scratch[8];
  const int n = blockIdx.x, i = threadIdx.x;

  float ix = 1.f / (1.f + __expf(-rowmax[n * HH + i]));
  float tx = rowor[n * HH + i];

  float cv = -3.4e38f, ty = 0.f;
  #pragma unroll
  for (int c = 0; c < NCH; c++) {
    cv = fmaxf(cv, colmax[((size_t)(n * NCH) + c) * WW + i]);
    ty = fmaxf(ty, color [((size_t)(n * NCH) + c) * WW + i]);
  }
  float iy = 1.f / (1.f + __expf(-cv));

  float s1 = block_sum_wmma(ix * tx, scratch);
  float s2 = block_sum_wmma(ix * ix, scratch);
  float s3 = block_sum_wmma(tx * tx, scratch);
  float t1 = block_sum_wmma(iy * ty, scratch);
  float t2 = block_sum_wmma(iy * iy, scratch);
  float t3 = block_sum_wmma(ty * ty, scratch);
  if (i == 0) {
    float lx = 1.f - 2.f * s1 / (s2 + s3 + 1e-3f);
    float ly = 1.f - 2.f * t1 / (t2 + t3 + 1e-3f);
    proj[n] = lx + ly;
  }
}

// --------------------------- K4: weighted cross-entropy ---------------------
__global__ __launch_bounds__(256) void k4_ce(
    const float* __restrict__ logits, const int* __restrict__ classes,
    const float* __restrict__ ew, float* __restrict__ ce) {
  __shared__ float scratch[8];
  const int t = threadIdx.x;
  float wn = 0.f, w = 0.f;
  if (t < 200) {
    const float* L = logits + (size_t)t * 81;
    float m = -3.4e38f;
    for (int c = 0; c < 81; c++) m = fmaxf(m, L[c]);
    float s = 0.f;
    for (int c = 0; c < 81; c++) s += __expf(L[c] - m);
    float logZ = m + __logf(s);
    int tc = classes[t];
    float nll = logZ - L[tc];
    w = ew[tc];
    wn = w * nll;
  }
  float a = block_sum_wmma(wn, scratch);
  float b = block_sum_wmma(w, scratch);
  if (t == 0) { ce[0] = a; ce[1] = b; }
}

// --------------------------- K5: deterministic combine ----------------------
__global__ __launch_bounds__(256) void k5_final(
    const float* __restrict__ pair_sim, const float* __restrict__ pair_cnt,
    const float* __restrict__ proj, const float* __restrict__ ce,
    const int* __restrict__ nmp, float* __restrict__ out) {
  __shared__ double sd[256];
  const int t = threadIdx.x;
  double ss = 0.0, sc = 0.0, sp = 0.0;
  for (int i = t; i < NN * HH; i += 256) { ss += pair_sim[i]; sc += pair_cnt[i]; }
  for (int i = t; i < NN; i += 256) sp += proj[i];

  auto red = [&](double v) -> double {
    sd[t] = v; __syncthreads();
    for (int o = 128; o > 0; o >>= 1) {
      if (t < o) sd[t] += sd[t + o];
      __syncthreads();
    }
    double r = sd[0]; __syncthreads();
    return r;
  };
  double S = red(ss), C = red(sc), P = red(sp);

  if (t == 0) {
    int nmi = *nmp; if (nmi < 1) nmi = 1;
    double nm = (double)nmi;
    float  lce = ce[0] / ce[1];
    double lp  = (S / (C > 1.0 ? C : 1.0)) / nm;
    double lpr = P / nm;
    out[0] = lce + (float)lp + (float)lpr;
  }
}

// ---------------------------------------------------------------------------
extern "C" void kernel_launch(void* const* d_in, const int* in_sizes, int n_in,
                              void* d_out, int out_size, void* d_ws, size_t ws_size,
                              hipStream_t stream) {
  (void)in_sizes; (void)n_in; (void)out_size; (void)ws_size;
  const float* logits  = (const float*)d_in[0];   // [2,100,81]
  const float* src     = (const float*)d_in[1];   // [64,256,256]
  const float* ew      = (const float*)d_in[2];   // [81]
  const float* ts      = (const float*)d_in[3];   // [64,8,256,256]
  const int*   classes = (const int*)  d_in[4];   // [2,100]
  const int*   tb      = (const int*)  d_in[5];   // [64,256,256]
  const int*   nmp     = (const int*)  d_in[6];   // scalar

  float* f = (float*)d_ws;
  float* pair_sim = f;                    // 16384
  float* pair_cnt = f + 16384;            // 16384
  float* rowmax   = f + 32768;            // 16384
  float* rowor    = f + 49152;            // 16384
  float* colmax   = f + 65536;            // 64*4*256 = 65536
  float* color    = f + 131072;           // 65536
  float* proj     = f + 196608;           // 64
  float* ce       = f + 196672;           // 2

  k1_pair_row<<<dim3(NN * HH), 256, 0, stream>>>(src, tb, ts, pair_sim, pair_cnt,
                                                 rowmax, rowor);
  k2_col<<<dim3(NN, NCH), 256, 0, stream>>>(src, tb, colmax, color);
  k3_proj<<<NN, 256, 0, stream>>>(rowmax, rowor, colmax, color, proj);
  k4_ce<<<1, 256, 0, stream>>>(logits, classes, ew, ce);
  k5_final<<<1, 256, 0, stream>>>(pair_sim, pair_cnt, proj, ce, nmp, (float*)d_out);
}